// AIModel_34892314312864
// MI455X (gfx1250) — compile-verified
//
#include <hip/hip_runtime.h>
#include <hip/hip_bf16.h>
#include <math.h>

typedef __attribute__((ext_vector_type(2))) float v2f;
typedef __attribute__((ext_vector_type(8))) float v8f;
typedef int i32x4_ __attribute__((vector_size(16)));   // matches builtin's pointee type
typedef __attribute__((address_space(1))) i32x4_* gvec4p;
typedef __attribute__((address_space(3))) i32x4_* lvec4p;

#define B_      512
#define D_      40
#define P_      30
#define M_      20
#define T_      8
#define V_      512
#define E_      64
#define MEDPAD  514
#define G3V     (3 * V_)

#if __has_builtin(__builtin_amdgcn_global_load_async_to_lds_b128)
#define HAVE_ASYNC_LDS 1
#else
#define HAVE_ASYNC_LDS 0
#endif

// 16-byte global -> LDS copy. Async DMA path on CDNA5 (ASYNCcnt-tracked),
// synchronous register copy otherwise.
__device__ __forceinline__ void cp16_g2l(float* lds, const float* g) {
#if HAVE_ASYNC_LDS
    __builtin_amdgcn_global_load_async_to_lds_b128(
        (gvec4p)(uintptr_t)g,
        (lvec4p)(uint32_t)(uintptr_t)lds,
        0, 0);
#else
    const float4 v = *reinterpret_cast<const float4*>(g);
    lds[0] = v.x; lds[1] = v.y; lds[2] = v.z; lds[3] = v.w;
#endif
}

__device__ __forceinline__ void wait_async_le4() {
#if HAVE_ASYNC_LDS
#if __has_builtin(__builtin_amdgcn_s_wait_asynccnt)
    __builtin_amdgcn_s_wait_asynccnt(4);
#else
    asm volatile("s_wait_asynccnt 0x4" ::: "memory");
#endif
#endif
}
__device__ __forceinline__ void wait_async_le0() {
#if HAVE_ASYNC_LDS
#if __has_builtin(__builtin_amdgcn_s_wait_asynccnt)
    __builtin_amdgcn_s_wait_asynccnt(0);
#else
    asm volatile("s_wait_asynccnt 0x0" ::: "memory");
#endif
#endif
}

// ---------------------------------------------------------------------------
// Embedding-bag pooling: fin[b, 0:64]=diag, [64:128]=proc, [128:192]=masked med
// ---------------------------------------------------------------------------
__global__ void pool_kernel(const int* __restrict__ diagnose,
                            const int* __restrict__ procedures,
                            const int* __restrict__ last_meds,
                            const float* __restrict__ diag_table,
                            const float* __restrict__ proc_table,
                            const float* __restrict__ med_table,
                            float* __restrict__ fin) {
    const int b = blockIdx.x;
    const int e = threadIdx.x;            // 0..63
    float s0 = 0.f, s1 = 0.f, s2 = 0.f;
    for (int d = 0; d < D_; ++d) s0 += diag_table[diagnose[b * D_ + d] * E_ + e];
    for (int p = 0; p < P_; ++p) s1 += proc_table[procedures[b * P_ + p] * E_ + e];
    for (int m = 0; m < M_; ++m) {
        const int idx = last_meds[b * M_ + m];
        if (idx != MEDPAD) s2 += med_table[idx * E_ + e];
    }
    fin[b * (3 * E_) + e]          = s0;
    fin[b * (3 * E_) + E_ + e]     = s1;
    fin[b * (3 * E_) + 2 * E_ + e] = s2;
}

// ---------------------------------------------------------------------------
// f32 WMMA GEMM:  C[M,N] = A[M,K] * B + bias
//   TRANS_B=1 : B[k,n] = W[n*ldb + k]  (X @ W.T; B tile staged [n][k])
//   TRANS_B=0 : B[k,n] = Bm[k*ldb + n] (B tile staged [k][n])
// 256 threads = 8 waves; block tile 64x64; wave tile 16x32 (2 accumulators,
// shared A fragment). K slabs of 32 double-buffered in LDS via async copies.
// Requires M%64==0, N%64==0, K%32==0 (true at all call sites).
// ---------------------------------------------------------------------------
template <bool TRANS_B>
__global__ __launch_bounds__(256) void wmma_gemm(
    const float* __restrict__ A, int lda,
    const float* __restrict__ Bm, int ldb,
    const float* __restrict__ bias,
    float* __restrict__ C, int ldc, int K) {
    constexpr int BM = 64, BN = 64, BK = 32;
    constexpr int APITCH = BK + 1;                   // 33
    constexpr int BROWS  = TRANS_B ? BN : BK;        // 64 : 32
    constexpr int BCOLS  = TRANS_B ? BK : BN;        // 32 : 64
    constexpr int BPITCH = BCOLS + 1;
    constexpr int BCPR   = BCOLS / 4;                // float4 chunks per B row

    __shared__ float As[2][BM][APITCH];
    __shared__ float Bs[2][BROWS][BPITCH];

    const int tid   = threadIdx.x;
    const int lane  = tid & 31;
    const int wave  = tid >> 5;
    const int m0    = blockIdx.y * BM;
    const int n0    = blockIdx.x * BN;
    const int m_loc = (wave & 3) * 16;
    const int n_loc = (wave >> 2) * 32;

    // Stage one K-slab (all copies are contiguous 16B moves).
    auto issue_slab = [&](int kt, int buf) {
#pragma unroll
        for (int i = 0; i < 2; ++i) {                 // A: 64x32 = 512 chunks
            const int q  = tid + i * 256;
            const int ra = q >> 3;
            const int ca = (q & 7) * 4;
            cp16_g2l(&As[buf][ra][ca], A + (size_t)(m0 + ra) * lda + kt + ca);
        }
#pragma unroll
        for (int i = 0; i < 2; ++i) {                 // B: 2048 floats = 512 chunks
            const int q  = tid + i * 256;
            const int rb = q / BCPR;
            const int cb = (q % BCPR) * 4;
            const float* src = TRANS_B
                ? (Bm + (size_t)(n0 + rb) * ldb + kt + cb)
                : (Bm + (size_t)(kt + rb) * ldb + n0 + cb);
            cp16_g2l(&Bs[buf][rb][cb], src);
        }
    };

    v8f acc0 = {0.f, 0.f, 0.f, 0.f, 0.f, 0.f, 0.f, 0.f};
    v8f acc1 = {0.f, 0.f, 0.f, 0.f, 0.f, 0.f, 0.f, 0.f};

    const int mrow  = m_loc + (lane & 15);
    const int ncol  = n_loc + (lane & 15);
    const int kh    = (lane >> 4) * 2;   // K half-select per ISA fragment layout

    const int nslabs = K / BK;
    issue_slab(0, 0);

    for (int s = 0; s < nslabs; ++s) {
        const int buf = s & 1;
        if (s + 1 < nslabs) {
            issue_slab((s + 1) * BK, buf ^ 1);
            wait_async_le4();             // in-order: slab s resident, s+1 in flight
        } else {
            wait_async_le0();
        }
        __syncthreads();

        // Gather all fragments for this slab, then issue WMMAs in a burst.
        v2f af[8], bf0[8], bf1[8];
#pragma unroll
        for (int u = 0; u < 8; ++u) {
            const int k = u * 4 + kh;
            af[u].x = As[buf][mrow][k];
            af[u].y = As[buf][mrow][k + 1];
            if (TRANS_B) {
                bf0[u].x = Bs[buf][ncol][k];        bf0[u].y = Bs[buf][ncol][k + 1];
                bf1[u].x = Bs[buf][ncol + 16][k];   bf1[u].y = Bs[buf][ncol + 16][k + 1];
            } else {
                bf0[u].x = Bs[buf][k][ncol];        bf0[u].y = Bs[buf][k + 1][ncol];
                bf1[u].x = Bs[buf][k][ncol + 16];   bf1[u].y = Bs[buf][k + 1][ncol + 16];
            }
        }
#pragma unroll
        for (int u = 0; u < 8; ++u) {
            acc0 = __builtin_amdgcn_wmma_f32_16x16x4_f32(
                false, af[u], false, bf0[u], (short)0, acc0, false, false);
            acc1 = __builtin_amdgcn_wmma_f32_16x16x4_f32(
                false, af[u], false, bf1[u], (short)0, acc1, false, false);
        }
        __syncthreads();
    }

    // Epilogue: C VGPR v -> row (m_base + v), col fixed per lane.
    const int na = n0 + ncol;
    const int nb = na + 16;
    const int mb = m0 + m_loc + ((lane >> 4) << 3);
    const float bva = bias ? bias[na] : 0.f;
    const float bvb = bias ? bias[nb] : 0.f;
#pragma unroll
    for (int v = 0; v < 8; ++v) {
        C[(size_t)(mb + v) * ldc + na] = acc0[v] + bva;
        C[(size_t)(mb + v) * ldc + nb] = acc1[v] + bvb;
    }
}

// ---------------------------------------------------------------------------
// Elementwise helpers
// ---------------------------------------------------------------------------
__global__ void zero_kernel(float* __restrict__ p, int n) {
    const int i = blockIdx.x * blockDim.x + threadIdx.x;
    if (i < n) p[i] = 0.f;
}

// GRU gate + masked state update (gate order r,z,n; h0=0). n-gate needs i_n
// and h_n separate, so gi/gh are distinct buffers. In-place h update is safe.
__global__ void gru_gate_kernel(const float* __restrict__ gi,
                                const float* __restrict__ gh,
                                const int* __restrict__ hist_len, int t,
                                float* __restrict__ h) {
    const int i = blockIdx.x * blockDim.x + threadIdx.x;
    if (i >= B_ * V_) return;
    const int b = i >> 9;
    const int v = i & (V_ - 1);
    const float* gib = gi + (size_t)b * G3V;
    const float* ghb = gh + (size_t)b * G3V;
    const float r  = 1.f / (1.f + expf(-(gib[v] + ghb[v])));
    const float z  = 1.f / (1.f + expf(-(gib[V_ + v] + ghb[V_ + v])));
    const float nn = tanhf(gib[2 * V_ + v] + r * ghb[2 * V_ + v]);
    const float hp = h[i];
    const float hn = (1.f - z) * nn + z * hp;
    h[i] = (t < hist_len[b]) ? hn : hp;
}

// out += h (GRU last hidden); cache S = sigmoid(out)
__global__ void add_sig_kernel(float* __restrict__ out,
                               const float* __restrict__ h,
                               float* __restrict__ S) {
    const int i = blockIdx.x * blockDim.x + threadIdx.x;
    if (i >= B_ * V_) return;
    const float o = out[i] + h[i];
    out[i] = o;
    S[i] = 1.f / (1.f + expf(-o));
}

// score = 1e-3 * 0.5 * mean_b sum_v tS[b,v]*S[b,v]; single-block reduce.
__global__ void ddi_reduce_kernel(const float* __restrict__ tS,
                                  const float* __restrict__ S,
                                  float* __restrict__ score) {
    __shared__ float red[256];
    float s = 0.f;
    for (int i = threadIdx.x; i < B_ * V_; i += 256) s += tS[i] * S[i];
    red[threadIdx.x] = s;
    __syncthreads();
    for (int w = 128; w > 0; w >>= 1) {
        if (threadIdx.x < w) red[threadIdx.x] += red[threadIdx.x + w];
        __syncthreads();
    }
    if (threadIdx.x == 0) *score = 0.001f * 0.5f * (red[0] / (float)B_);
}

// ---------------------------------------------------------------------------
extern "C" void kernel_launch(void* const* d_in, const int* in_sizes, int n_in,
                              void* d_out, int out_size, void* d_ws, size_t ws_size,
                              hipStream_t stream) {
    const int*   diagnose   = (const int*)  d_in[0];
    const int*   procedures = (const int*)  d_in[1];
    const int*   last_meds  = (const int*)  d_in[2];
    const float* med_hist   = (const float*)d_in[3];
    const int*   hist_len   = (const int*)  d_in[4];
    const float* ddi_adj    = (const float*)d_in[5];
    const float* diag_table = (const float*)d_in[6];
    const float* proc_table = (const float*)d_in[7];
    const float* med_table  = (const float*)d_in[8];
    const float* W_out      = (const float*)d_in[9];
    const float* b_out      = (const float*)d_in[10];
    const float* W_ih       = (const float*)d_in[11];
    const float* W_hh       = (const float*)d_in[12];
    const float* b_ih       = (const float*)d_in[13];
    const float* b_hh       = (const float*)d_in[14];

    float* out = (float*)d_out;          // [B*V] logits+h, score at [B*V]
    float* ws  = (float*)d_ws;
    float* fin = ws;                     // B*192
    float* h   = fin + B_ * 3 * E_;      // B*V
    float* gi  = h + B_ * V_;            // B*3V
    float* gh  = gi + (size_t)B_ * G3V;  // B*3V
    float* S   = gh + (size_t)B_ * G3V;  // B*V
    float* tS  = S + B_ * V_;            // B*V

    const int BV = B_ * V_;
    const dim3 blk256(256);

    // 1) embedding pools -> fin [B, 192]
    pool_kernel<<<B_, 64, 0, stream>>>(diagnose, procedures, last_meds,
                                       diag_table, proc_table, med_table, fin);

    // 2) logits = fin @ W_out.T + b_out -> out (M=N=512, K=192)
    wmma_gemm<true><<<dim3(V_ / 64, B_ / 64), blk256, 0, stream>>>(
        fin, 3 * E_, W_out, 3 * E_, b_out, out, V_, 3 * E_);

    // 3) GRU: h = 0; 8 masked steps of two 512x1536x512 WMMA GEMMs + gates
    zero_kernel<<<(BV + 255) / 256, 256, 0, stream>>>(h, BV);
    for (int t = 0; t < T_; ++t) {
        wmma_gemm<true><<<dim3(G3V / 64, B_ / 64), blk256, 0, stream>>>(
            med_hist + t * V_, T_ * V_, W_ih, V_, b_ih, gi, G3V, V_);
        wmma_gemm<true><<<dim3(G3V / 64, B_ / 64), blk256, 0, stream>>>(
            h, V_, W_hh, V_, b_hh, gh, G3V, V_);
        gru_gate_kernel<<<(BV + 255) / 256, 256, 0, stream>>>(gi, gh, hist_len, t, h);
    }

    // 4) out = logits + h ; S = sigmoid(out)
    add_sig_kernel<<<(BV + 255) / 256, 256, 0, stream>>>(out, h, S);

    // 5) tS = S @ ddi_adj (symmetric => einsum('bi,ij->bj'))
    wmma_gemm<false><<<dim3(V_ / 64, B_ / 64), blk256, 0, stream>>>(
        S, V_, ddi_adj, V_, nullptr, tS, V_, V_);

    // 6) score = 5e-4 * mean_b <tS_b, S_b>
    ddi_reduce_kernel<<<1, 256, 0, stream>>>(tS, S, out + BV);
}